// Memory_29884382445748
// MI455X (gfx1250) — compile-verified
//
#include <hip/hip_runtime.h>
#include <math.h>

typedef float v2f __attribute__((ext_vector_type(2)));
typedef float v8f __attribute__((ext_vector_type(8)));
typedef int   gv4i __attribute__((vector_size(16)));   // matches builtin param pointee

#define B_   4
#define De_  128
#define Do_  512
#define T_   8
#define H_   30
#define W_   54
#define THW_ (T_*H_*W_)      // 12960
#define HW_  (H_*W_)         // 1620
#define NT_  ((HW_+15)/16)   // 102 column tiles (last partial: 4 cols)
#define NCHUNK_ 40
#define ROWS_PER_CHUNK_ (THW_/NCHUNK_)  // 324
#define BHW_ (B_*HW_)        // 6480

// GEMM2 tiling
#define KB_     96                       // K-panel rows staged in LDS
#define NPANEL_ (THW_/KB_)               // 135
#define NCOLT_  4                        // N-tiles (16 cols each) per wave
#define NTG_    ((NT_+NCOLT_-1)/NCOLT_)  // 26 column groups of 64
#define LDSW_   68                       // padded row stride (floats) -> conflict-free

// ---------------------------------------------------------------------------
// gfx1250 async global->LDS copy (ASYNCcnt) with guaranteed-compile fallback.
// Builtin signature (from hipcc diagnostic): params are pointers to
// 16-byte GCC int vectors; generic pointers implicitly convert in HIP.
// ---------------------------------------------------------------------------
#if defined(__has_builtin)
# if __has_builtin(__builtin_amdgcn_global_load_async_to_lds_b128)
#   define HAVE_ASYNC_B128 1
# endif
#endif

__device__ __forceinline__ void async_g2l_b128(const float* g, float* l) {
#ifdef HAVE_ASYNC_B128
    __builtin_amdgcn_global_load_async_to_lds_b128((gv4i*)g, (gv4i*)l, 0, 0);
#else
    *(float4*)l = *(const float4*)g;     // synchronous fallback
#endif
}

__device__ __forceinline__ void wait_async0() {
#ifdef HAVE_ASYNC_B128
# if defined(__has_builtin) && __has_builtin(__builtin_amdgcn_s_wait_asynccnt)
    __builtin_amdgcn_s_wait_asynccnt(0);
# else
    asm volatile("s_wait_asynccnt 0x0" ::: "memory");
# endif
#endif
}

// ---------------------------------------------------------------------------
// GEMM1: p_raw[b, m, q] = (1/sqrt(De)) * sum_d mi[b,d,m] * qi[b,d,q]
// One wave computes one 16x16 tile via V_WMMA_F32_16X16X4_F32.
// ---------------------------------------------------------------------------
__global__ void __launch_bounds__(256)
gemm_p_kernel(const float* __restrict__ mi,
              const float* __restrict__ qi,
              float* __restrict__ p_raw) {
    const int lane = threadIdx.x & 31;
    const int wave = threadIdx.x >> 5;
    const int tile = blockIdx.x * 8 + wave;
    const int mtiles = THW_ / 16;            // 810
    const int b  = tile / (mtiles * NT_);
    const int r  = tile % (mtiles * NT_);
    const int mt = r / NT_;
    const int nt = r % NT_;
    const int half = lane >> 4;
    const int lm   = lane & 15;
    const int m = mt * 16 + lm;
    const int n = nt * 16 + lm;
    const int nc = (n < HW_) ? n : (HW_ - 1);   // clamp loads; store guarded

    const float* A  = mi + (size_t)b * De_ * THW_;
    const float* Bp = qi + (size_t)b * De_ * HW_;

    v8f acc = {};
    #pragma unroll 4
    for (int k0 = 0; k0 < De_; k0 += 4) {
        const int ka = k0 + 2 * half;
        v2f af, bf;
        af.x = A[(size_t)ka       * THW_ + m];
        af.y = A[(size_t)(ka + 1) * THW_ + m];
        bf.x = Bp[(size_t)ka       * HW_ + nc];
        bf.y = Bp[(size_t)(ka + 1) * HW_ + nc];
        acc = __builtin_amdgcn_wmma_f32_16x16x4_f32(false, af, false, bf,
                                                    (short)0, acc, false, false);
    }

    if (n < HW_) {
        const float scale = 0.0883883476483184f;  // 1/sqrt(128)
        float* out = p_raw + (size_t)b * THW_ * HW_;
        #pragma unroll
        for (int rr = 0; rr < 8; ++rr) {
            const int row = mt * 16 + rr + 8 * half;
            out[(size_t)row * HW_ + n] = acc[rr] * scale;
        }
    }
}

// ---------------------------------------------------------------------------
// Row argmax over HW per (b,m) row (first-max semantics like jnp.argmax).
// ---------------------------------------------------------------------------
__global__ void argmax_kernel(const float* __restrict__ p_raw,
                              int* __restrict__ idx_out) {
    __shared__ float smax[128];
    __shared__ int   sidx[128];
    const int row = blockIdx.x;                 // b*THW + m
    const float* rp = p_raw + (size_t)row * HW_;
    float best = -INFINITY;
    int   bi   = 0x7fffffff;
    for (int q = threadIdx.x; q < HW_; q += 128) {
        const float v = rp[q];
        if (v > best) { best = v; bi = q; }     // ascending q -> first max kept
    }
    smax[threadIdx.x] = best;
    sidx[threadIdx.x] = bi;
    __syncthreads();
    for (int s = 64; s > 0; s >>= 1) {
        if (threadIdx.x < s) {
            const float ov = smax[threadIdx.x + s];
            const int   oi = sidx[threadIdx.x + s];
            if (ov > smax[threadIdx.x] ||
                (ov == smax[threadIdx.x] && oi < sidx[threadIdx.x])) {
                smax[threadIdx.x] = ov;
                sidx[threadIdx.x] = oi;
            }
        }
        __syncthreads();
    }
    if (threadIdx.x == 0) idx_out[row] = sidx[0];
}

// ---------------------------------------------------------------------------
// Column max partials over THW (softmax is over the memory axis).
// ---------------------------------------------------------------------------
__global__ void colmax_partial(const float* __restrict__ p_raw,
                               float* __restrict__ pmax) {
    const int bq = blockIdx.x * blockDim.x + threadIdx.x;
    if (bq >= BHW_) return;
    const int chunk = blockIdx.y;
    const int b = bq / HW_, q = bq % HW_;
    const float* base = p_raw + ((size_t)b * THW_ + (size_t)chunk * ROWS_PER_CHUNK_) * HW_ + q;
    float mx = -INFINITY;
    for (int m = 0; m < ROWS_PER_CHUNK_; ++m)
        mx = fmaxf(mx, base[(size_t)m * HW_]);
    pmax[(size_t)chunk * BHW_ + bq] = mx;
}

__global__ void colmax_reduce(const float* __restrict__ pmax,
                              float* __restrict__ colmax) {
    const int bq = blockIdx.x * blockDim.x + threadIdx.x;
    if (bq >= BHW_) return;
    float mx = -INFINITY;
    for (int c = 0; c < NCHUNK_; ++c)
        mx = fmaxf(mx, pmax[(size_t)c * BHW_ + bq]);
    colmax[bq] = mx;
}

// ---------------------------------------------------------------------------
// In-place: w = exp(p - colmax) * gauss(argmax_idx[m], q); chunked col sums.
// ---------------------------------------------------------------------------
__global__ void weight_partial(float* __restrict__ p,
                               const int* __restrict__ idx,
                               const float* __restrict__ colmax,
                               float* __restrict__ psum) {
    const int bq = blockIdx.x * blockDim.x + threadIdx.x;
    if (bq >= BHW_) return;
    const int chunk = blockIdx.y;
    const int b = bq / HW_, q = bq % HW_;
    const float qy = (float)(q / W_);
    const float qx = (float)(q % W_);
    const float cm = colmax[bq];
    const int m0 = chunk * ROWS_PER_CHUNK_;
    float* base = p + ((size_t)b * THW_ + m0) * HW_ + q;
    const int* ib = idx + b * THW_ + m0;
    const float inv2s2 = 1.0f / 18.0f;   // sigma = 3
    float s = 0.0f;
    for (int m = 0; m < ROWS_PER_CHUNK_; ++m) {
        const int id = ib[m];
        const float iy = (float)(id / W_);
        const float ix = (float)(id % W_);
        const float dx = qx - ix, dy = qy - iy;
        const float g = __expf(-(dx * dx + dy * dy) * inv2s2);
        const float w = __expf(base[(size_t)m * HW_] - cm) * g;
        base[(size_t)m * HW_] = w;
        s += w;
    }
    psum[(size_t)chunk * BHW_ + bq] = s;
}

__global__ void colsum_reduce(const float* __restrict__ psum,
                              float* __restrict__ rsum) {
    const int bq = blockIdx.x * blockDim.x + threadIdx.x;
    if (bq >= BHW_) return;
    float s = 0.0f;
    for (int c = 0; c < NCHUNK_; ++c)
        s += psum[(size_t)c * BHW_ + bq];
    rsum[bq] = 1.0f / s;
}

__global__ void normalize_kernel(float* __restrict__ p,
                                 const float* __restrict__ rsum) {
    const size_t e = (size_t)blockIdx.x * 256 + threadIdx.x;
    if (e >= (size_t)B_ * THW_ * HW_) return;
    const int q = (int)(e % HW_);
    const int b = (int)(e / ((size_t)THW_ * HW_));
    p[e] *= rsum[b * HW_ + q];
}

// ---------------------------------------------------------------------------
// GEMM2: mem[b, do, q] = sum_m mo[b,do,m] * p[b,m,q]
// Block = 8 waves; each wave: one M-tile x four N-tiles (A-fragment reuse 4x).
// B panels (KB_ x 64 f32) staged in LDS via async global->LDS, double-buffered.
// ---------------------------------------------------------------------------
__global__ void __launch_bounds__(256)
gemm_mem_kernel(const float* __restrict__ mo,
                const float* __restrict__ p,
                float* __restrict__ out) {
    __shared__ float lds[2][KB_ * LDSW_];

    const int lane = threadIdx.x & 31;
    const int wave = threadIdx.x >> 5;
    const int blk  = blockIdx.x;             // b*(4*NTG_) + mg*NTG_ + ntg
    const int b    = blk / (4 * NTG_);
    const int rr0  = blk % (4 * NTG_);
    const int mg   = rr0 / NTG_;
    const int ntg  = rr0 % NTG_;
    const int half = lane >> 4;
    const int lm   = lane & 15;
    const int mt   = mg * 8 + wave;
    const int m    = mt * 16 + lm;
    const int col0 = ntg * 64;

    const float* A  = mo + (size_t)b * Do_ * THW_ + (size_t)m * THW_;
    const float* Bp = p  + (size_t)b * THW_ * HW_;

    // Stage panel pi of p (KB_ rows x 64 cols) into lds[buf]: 16B chunks.
    auto issue = [&](int pi, int buf) {
        const int k0g = pi * KB_;
        #pragma unroll
        for (int i = 0; i < (KB_ * 16) / 256; ++i) {      // 6 chunks/thread
            const int chd = threadIdx.x + i * 256;
            const int k   = chd >> 4;
            const int c4  = (chd & 15) * 4;
            int gc = col0 + c4;
            if (gc > HW_ - 4) gc = HW_ - 4;               // aligned clamp
            async_g2l_b128(Bp + (size_t)(k0g + k) * HW_ + gc,
                           &lds[buf][k * LDSW_ + c4]);
        }
    };

    v8f acc0 = {}, acc1 = {}, acc2 = {}, acc3 = {};
    issue(0, 0);
    for (int pi = 0; pi < NPANEL_; ++pi) {
        const int buf = pi & 1;
        wait_async0();
        __syncthreads();   // panel ready; all waves done reading buf^1
        if (pi + 1 < NPANEL_) issue(pi + 1, buf ^ 1);

        const float* Ap = A + pi * KB_;
        const float* L  = lds[buf];
        #pragma unroll 4
        for (int kk = 0; kk < KB_; kk += 4) {
            const int ka = kk + 2 * half;
            const v2f af = *(const v2f*)(Ap + ka);        // 8B-aligned (ka even)
            const float* L0 = L + ka * LDSW_ + lm;
            const float* L1 = L0 + LDSW_;
            v2f bf;
            bf.x = L0[0];  bf.y = L1[0];
            acc0 = __builtin_amdgcn_wmma_f32_16x16x4_f32(false, af, false, bf,
                                                         (short)0, acc0, false, false);
            bf.x = L0[16]; bf.y = L1[16];
            acc1 = __builtin_amdgcn_wmma_f32_16x16x4_f32(false, af, false, bf,
                                                         (short)0, acc1, false, false);
            bf.x = L0[32]; bf.y = L1[32];
            acc2 = __builtin_amdgcn_wmma_f32_16x16x4_f32(false, af, false, bf,
                                                         (short)0, acc2, false, false);
            bf.x = L0[48]; bf.y = L1[48];
            acc3 = __builtin_amdgcn_wmma_f32_16x16x4_f32(false, af, false, bf,
                                                         (short)0, acc3, false, false);
        }
        __syncthreads();
    }

    float* outb = out + (size_t)b * (2 * Do_) * HW_;
    #pragma unroll
    for (int j = 0; j < 4; ++j) {
        const v8f acc = (j == 0) ? acc0 : (j == 1) ? acc1 : (j == 2) ? acc2 : acc3;
        const int col = col0 + j * 16 + lm;
        if (col < HW_) {
            #pragma unroll
            for (int rr = 0; rr < 8; ++rr) {
                const int row = mt * 16 + rr + 8 * half;
                outb[(size_t)row * HW_ + col] = acc[rr];
            }
        }
    }
}

// ---------------------------------------------------------------------------
// Copy q_out into channels Do..2Do-1 of mem_out.
// ---------------------------------------------------------------------------
__global__ void concat_qout(const float* __restrict__ q_out,
                            float* __restrict__ out) {
    const size_t per_b = (size_t)Do_ * HW_;
    const size_t e = (size_t)blockIdx.x * 256 + threadIdx.x;
    if (e >= (size_t)B_ * per_b) return;
    const int b = (int)(e / per_b);
    const size_t off = e % per_b;
    out[(size_t)b * (2 * Do_) * HW_ + per_b + off] = q_out[e];
}

// ---------------------------------------------------------------------------
extern "C" void kernel_launch(void* const* d_in, const int* in_sizes, int n_in,
                              void* d_out, int out_size, void* d_ws, size_t ws_size,
                              hipStream_t stream) {
    const float* mi = (const float*)d_in[0];   // m_in  (B, De, T, H, W)
    const float* mo = (const float*)d_in[1];   // m_out (B, Do, T, H, W)
    const float* qi = (const float*)d_in[2];   // q_in  (B, De, H, W)
    const float* qo = (const float*)d_in[3];   // q_out (B, Do, H, W)
    float* out = (float*)d_out;

    // Output layout: [mem_out (B,2Do,H,W)][p (B,THW,HW)]
    const size_t mem_elems = (size_t)B_ * 2 * Do_ * HW_;   // 6,635,520
    float* out_p = out + mem_elems;                        // p lives here end-to-end

    // Workspace partition (aligned to 256B), total ~2.3 MB.
    char* ws = (char*)d_ws;
    size_t off = 0;
    auto take = [&](size_t bytes) { char* pp = ws + off; off += (bytes + 255) & ~(size_t)255; return pp; };
    int*   idx    = (int*)  take((size_t)B_ * THW_ * sizeof(int));
    float* pmax   = (float*)take((size_t)NCHUNK_ * BHW_ * sizeof(float));
    float* colmax = (float*)take((size_t)BHW_ * sizeof(float));
    float* psum   = (float*)take((size_t)NCHUNK_ * BHW_ * sizeof(float));
    float* rsum   = (float*)take((size_t)BHW_ * sizeof(float));
    (void)ws_size; (void)in_sizes; (void)n_in; (void)out_size;

    const int cols_blocks = (BHW_ + 255) / 256;            // 26

    // 1) p_raw = mi^T qi / sqrt(De)  (WMMA) -> out_p
    {
        const int tiles = B_ * (THW_ / 16) * NT_;          // 330,480
        gemm_p_kernel<<<tiles / 8, 256, 0, stream>>>(mi, qi, out_p);
    }
    // 2) per-row argmax
    argmax_kernel<<<B_ * THW_, 128, 0, stream>>>(out_p, idx);
    // 3) column max (chunked)
    colmax_partial<<<dim3(cols_blocks, NCHUNK_), 256, 0, stream>>>(out_p, pmax);
    colmax_reduce<<<cols_blocks, 256, 0, stream>>>(pmax, colmax);
    // 4) w = exp(p - max) * gauss, in place; chunked column sums
    weight_partial<<<dim3(cols_blocks, NCHUNK_), 256, 0, stream>>>(out_p, idx, colmax, psum);
    colsum_reduce<<<cols_blocks, 256, 0, stream>>>(psum, rsum);
    // 5) normalize p in place (final p output)
    {
        const size_t total = (size_t)B_ * THW_ * HW_;      // 83,980,800
        normalize_kernel<<<(unsigned)(total / 256), 256, 0, stream>>>(out_p, rsum);
    }
    // 6) mem = mo @ p  (WMMA, LDS-staged async B, 4 N-tiles/wave)
    gemm_mem_kernel<<<B_ * 4 * NTG_, 256, 0, stream>>>(mo, out_p, out);
    // 7) concat q_out -> channels Do..2Do-1
    {
        const size_t total = (size_t)B_ * Do_ * HW_;       // 3,317,760
        concat_qout<<<(unsigned)(total / 256), 256, 0, stream>>>(qo, out);
    }
}